// MXIntGELU_61314953118471
// MI455X (gfx1250) — compile-verified
//
#include <hip/hip_runtime.h>
#include <stdint.h>

// ---------------------------------------------------------------------------
// MXINT8 GELU (block = 1x32, W=8/EW=4 in & out, 8-bit hash GELU) for gfx1250.
// Bandwidth-bound streaming kernel: 268 MB traffic -> ~11.5us floor @23.3TB/s.
// Data path: global -> LDS via GLOBAL_LOAD_ASYNC_TO_LDS_B128 (ASYNCcnt,
// 4-stage x 2-slot pipeline = 8 B128 loads in flight per wave, th:NT),
// compute in VGPRs, nontemporal 128-bit stores. One MX block (32 floats) =
// 8 wave32 lanes x float4; block amax via shfl_xor(1,2,4).
// Contiguous chunk ownership: pure 32-bit address math, no device division.
// ---------------------------------------------------------------------------

typedef float v4f __attribute__((ext_vector_type(4)));

#define BLOCK    256
#define PER      2                        // float4 per thread per stage
#define STAGES   4
#define CHUNK_F4 (BLOCK * PER)            // 512 float4 = 8KB per chunk
#define SLOT_B   (BLOCK * 16u)            // bytes per slot
#define STAGE_B  (CHUNK_F4 * 16u)         // bytes per stage

__device__ __forceinline__ void async_load_f4(uint32_t lds_addr,
                                              const void* gbase,
                                              uint32_t byte_off) {
  // GVS mode: mem_addr = SGPR64 base + VGPR32 byte offset; NT streaming hint
  asm volatile("global_load_async_to_lds_b128 %0, %1, %2 th:TH_LOAD_NT"
               :: "v"(lds_addr), "v"(byte_off), "s"(gbase)
               : "memory");
}

__device__ __forceinline__ void wait_async_oldest_stage(void) {
  // 2 loads/stage, 4 stages in flight: <= 6 outstanding => oldest stage done
  asm volatile("s_wait_asynccnt 0x6" ::: "memory");
}

__device__ __forceinline__ float group8_max(float v) {
  // reduce max over the 8-lane group owning one 32-element MX block
  v = fmaxf(v, __shfl_xor(v, 1, 32));
  v = fmaxf(v, __shfl_xor(v, 2, 32));
  v = fmaxf(v, __shfl_xor(v, 4, 32));
  return v;
}

// branchless Abramowitz-Stegun 7.1.26 erf (|err| < 1.5e-7; only used |x|<3)
__device__ __forceinline__ float erf_fast(float x) {
  float z = fabsf(x);
  float t = __builtin_amdgcn_rcpf(fmaf(0.3275911f, z, 1.0f));
  float p = t * fmaf(t, fmaf(t, fmaf(t, fmaf(t, 1.061405429f, -1.453152027f),
                                     1.421413741f), -0.284496736f),
                     0.254829592f);
  float e = __expf(-z * z);               // v_exp_f32
  return copysignf(1.0f - p * e, x);
}

__device__ __forceinline__ void block_scales(float amax, float& sc, float& inv) {
  // ex = clip(floor(log2(max(amax, 2^-8))), -8, 7); sc = 2^(7-ex); inv = 2^(ex-7)
  float v = fmaxf(amax, 0.00390625f);                       // 2^-8
  int e = (int)((__float_as_uint(v) >> 23) & 0xFFu) - 127;  // floor(log2) >= -8
  e = e > 7 ? 7 : e;
  sc  = __uint_as_float((uint32_t)(134 - e) << 23);         // 2^(7-e)
  inv = __uint_as_float((uint32_t)(120 + e) << 23);         // 2^(e-7)
}

// full MXINT8 -> hashGELU -> MXINT8 path for one float4 (one 8-lane block)
__device__ __forceinline__ v4f process_f4(v4f x) {
  float amax = fmaxf(fmaxf(fabsf(x[0]), fabsf(x[1])),
                     fmaxf(fabsf(x[2]), fabsf(x[3])));
  amax = group8_max(amax);
  float sc, inv;
  block_scales(amax, sc, inv);

  v4f q;
#pragma unroll
  for (int j = 0; j < 4; ++j) {
    float m = rintf(x[j] * sc);                       // RTNE == jnp.round
    m = fminf(fmaxf(m, -128.0f), 127.0f);
    float qx = m * inv;                               // dequantized input
    float r  = fmaxf(qx, 0.0f);                       // relu branch
    float g  = 0.5f * qx * (1.0f + erf_fast(qx * 0.7071067811865476f));
    float qg = fminf(fmaxf(rintf(g * 32.0f), -128.0f), 127.0f) * 0.03125f;
    float qg2 = floorf(qg * sc) * inv;                // exponent-aligned requant
    bool rem = (qx > -3.0f) && (qx < 3.0f);
    q[j] = rem ? qg2 : r;
  }

  float a2 = fmaxf(fmaxf(fabsf(q[0]), fabsf(q[1])),
                   fmaxf(fabsf(q[2]), fabsf(q[3])));
  a2 = group8_max(a2);
  float sc2, inv2;
  block_scales(a2, sc2, inv2);

  v4f y;
#pragma unroll
  for (int j = 0; j < 4; ++j) {
    float m2 = fminf(fmaxf(rintf(q[j] * sc2), -128.0f), 127.0f);
    y[j] = m2 * inv2;
  }
  return y;
}

__global__ __launch_bounds__(BLOCK)
void mxint_gelu_kernel(const v4f* __restrict__ in, v4f* __restrict__ out,
                       unsigned total4, unsigned nchunks, unsigned ipb) {
  __shared__ v4f smem[STAGES][PER][BLOCK];
  const unsigned tid  = threadIdx.x;
  const unsigned base = blockIdx.x * ipb;              // first owned chunk
  if (base >= nchunks) return;
  const unsigned rem  = nchunks - base;
  const unsigned n_i  = rem < ipb ? rem : ipb;

  // low 32 bits of a generic LDS pointer == LDS byte offset on gfx1250
  const uint32_t lds0 = (uint32_t)(uintptr_t)(&smem[0][0][tid]);

  // ---- prologue: fill the pipeline (dummy-reload base chunk at the tail) --
#pragma unroll
  for (unsigned p = 0; p < STAGES; ++p) {
    unsigned c = base + p;
    if (c >= nchunks) c = base;
    unsigned i0 = c * CHUNK_F4 + tid;                  // slot 0 float4 index
    unsigned i1 = i0 + BLOCK;                          // slot 1 float4 index
    if (i0 >= total4) i0 = 0;
    if (i1 >= total4) i1 = 0;
    async_load_f4(lds0 + p * STAGE_B,          (const void*)in, i0 * 16u);
    async_load_f4(lds0 + p * STAGE_B + SLOT_B, (const void*)in, i1 * 16u);
  }

  // ---- steady-state loop --------------------------------------------------
  for (unsigned i = 0; i < n_i; ++i) {
    const unsigned stage = i & (STAGES - 1u);
    wait_async_oldest_stage();                         // stage i loads done
    v4f x0 = smem[stage][0][tid];
    v4f x1 = smem[stage][1][tid];

    v4f y0 = process_f4(x0);
    v4f y1 = process_f4(x1);

    const unsigned c  = base + i;
    const unsigned i0 = c * CHUNK_F4 + tid;
    const unsigned i1 = i0 + BLOCK;
    if (i0 < total4) __builtin_nontemporal_store(y0, out + i0);
    if (i1 < total4) __builtin_nontemporal_store(y1, out + i1);

    // re-issue this stage for iteration i+STAGES (dummy at the tail)
    unsigned cn = c + STAGES;
    if (cn >= nchunks) cn = base;
    unsigned n0 = cn * CHUNK_F4 + tid;
    unsigned n1 = n0 + BLOCK;
    if (n0 >= total4) n0 = 0;
    if (n1 >= total4) n1 = 0;
    async_load_f4(lds0 + stage * STAGE_B,          (const void*)in, n0 * 16u);
    async_load_f4(lds0 + stage * STAGE_B + SLOT_B, (const void*)in, n1 * 16u);
  }
}

extern "C" void kernel_launch(void* const* d_in, const int* in_sizes, int n_in,
                              void* d_out, int out_size, void* d_ws,
                              size_t ws_size, hipStream_t stream) {
  (void)n_in; (void)d_ws; (void)ws_size; (void)out_size;
  const v4f* x = (const v4f*)d_in[0];
  v4f* y       = (v4f*)d_out;
  const unsigned n       = (unsigned)in_sizes[0];     // 4*2048*4096
  const unsigned total4  = n / 4u;
  const unsigned nchunks = (total4 + CHUNK_F4 - 1u) / CHUNK_F4;   // 16384
  const unsigned target  = 2048u;
  const unsigned ipb     = (nchunks + target - 1u) / target;      // chunks/WG
  const unsigned grid    = (nchunks + ipb - 1u) / ipb;
  hipLaunchKernelGGL(mxint_gelu_kernel, dim3(grid), dim3(BLOCK), 0, stream,
                     x, y, total4, nchunks, ipb);
}